// ODEVAE_60988535603646
// MI455X (gfx1250) — compile-verified
//
#include <hip/hip_runtime.h>
#include <hip/hip_bf16.h>
#include <math.h>

// Latent ODE-VAE forward for MI455X (gfx1250), fp32 WMMA (V_WMMA_F32_16X16X4_F32).
// B=256, L=200, N=256, D=16, K=8 dopri substeps.
//
// ODE kernel: 16 persistent workgroups (one per 16-row batch tile), 8 wave32s.
// All ODE weight fragments live in VGPRs for the whole 199*8*6 = 9552-step chain;
// only activations cross waves through LDS. Layernorm stats via 16-lane shfl_xor.

typedef float v2f __attribute__((ext_vector_type(2)));
typedef float v8f __attribute__((ext_vector_type(8)));

#define B_ 256
#define L_ 200
#define N_ 256
#define D_ 16
#define KSUB_ 8
#define LN_EPS_ 1e-5f
#define ROWT_ 4  // decoder row-tiles (of 16) per workgroup

// Dormand-Prince coefficients (folded to immediates by full unrolling)
constexpr float RK_A[6][5] = {
    {0.f, 0.f, 0.f, 0.f, 0.f},
    {0.2f, 0.f, 0.f, 0.f, 0.f},
    {3.f/40.f, 9.f/40.f, 0.f, 0.f, 0.f},
    {44.f/45.f, -56.f/15.f, 32.f/9.f, 0.f, 0.f},
    {19372.f/6561.f, -25360.f/2187.f, 64448.f/6561.f, -212.f/729.f, 0.f},
    {9017.f/3168.f, -355.f/33.f, 46732.f/5247.f, 49.f/176.f, -5103.f/18656.f}};
constexpr float RK_B[6] = {35.f/384.f, 0.f, 500.f/1113.f, 125.f/192.f,
                           -2187.f/6784.f, 11.f/84.f};

// D = A*B + C, fp32 16x16x4. Fragments per CDNA5 ISA 7.12.2 (r=lane&15, half=lane>>4):
//   A frag: {A[r][k0+2h], A[r][k0+2h+1]}   B frag: {B[k0+2h][n0+r], B[k0+2h+1][n0+r]}
//   C frag: vgpr i -> row (i+8h), col (n0+r)
__device__ __forceinline__ v8f wmma_acc(v2f a, v2f b, v8f c) {
  return __builtin_amdgcn_wmma_f32_16x16x4_f32(false, a, false, b, (short)0, c,
                                               false, false);
}

// Pointer-based variant (A row-major lda from LDS, B row-major ldb from global/LDS)
__device__ __forceinline__ v8f wmma4(const float* __restrict__ A, int lda, int k0,
                                     const float* __restrict__ Bm, int ldb, int n0,
                                     v8f c, int r, int half) {
  const int kk = k0 + 2 * half;
  v2f a = *(const v2f*)(A + r * lda + kk);
  v2f b;
  b.x = Bm[kk * ldb + n0 + r];
  b.y = Bm[(kk + 1) * ldb + n0 + r];
  return wmma_acc(a, b, c);
}

// fast silu: v_exp_f32 + v_rcp_f32 path (~1e-6 rel err; layernorm re-normalizes
// every RK stage so the sequential chain stays bounded)
__device__ __forceinline__ float silu_f(float v) {
  return v * __builtin_amdgcn_rcpf(1.0f + __expf(-v));
}

__launch_bounds__(256) __global__
void ode_kernel(const float* __restrict__ tvec, const float* __restrict__ ow1,
                const float* __restrict__ ob1, const float* __restrict__ ow2,
                const float* __restrict__ ob2, const float* __restrict__ ow3,
                const float* __restrict__ ob3, const float* __restrict__ ln_g,
                const float* __restrict__ ln_b, const float* __restrict__ z0buf,
                float* __restrict__ ztraj) {
  extern __shared__ float smem[];
  float* zs = smem;         // 256
  float* a1 = zs + 256;     // 16*128
  float* a2 = a1 + 2048;    // 16*128
  float* part = a2 + 2048;  // 8*256

  const int g = blockIdx.x;
  const int tid = threadIdx.x;
  const int w = tid >> 5, lane = tid & 31, half = lane >> 4, r = lane & 15;
  const int n0 = w * 16;
  const int row = tid >> 4, col = tid & 15;

  // ---- weight B-fragments resident in VGPRs for the whole integration ----
  v2f w1f[4], w2f[32], w3f[4];
#pragma unroll
  for (int c2 = 0; c2 < 4; ++c2) {
    const int kk = 4 * c2 + 2 * half;
    w1f[c2].x = ow1[kk * 128 + n0 + r];
    w1f[c2].y = ow1[(kk + 1) * 128 + n0 + r];
  }
#pragma unroll
  for (int c2 = 0; c2 < 32; ++c2) {
    const int kk = 4 * c2 + 2 * half;
    w2f[c2].x = ow2[kk * 128 + n0 + r];
    w2f[c2].y = ow2[(kk + 1) * 128 + n0 + r];
  }
#pragma unroll
  for (int c2 = 0; c2 < 4; ++c2) {  // layer-3: wave w covers K in [16w,16w+16)
    const int kk = n0 + 4 * c2 + 2 * half;
    w3f[c2].x = ow3[kk * 16 + r];
    w3f[c2].y = ow3[(kk + 1) * 16 + r];
  }
  const float bias1 = ob1[n0 + r];
  const float bias2 = ob2[n0 + r];
  const float bias3 = ob3[col];
  const float gam = ln_g[col];
  const float bet = ln_b[col];

  float zreg = z0buf[g * 256 + tid];  // thread t owns z element (row,col)
  float kreg[6];

  for (int iv = 0; iv < L_ - 1; ++iv) {
    const float h = (tvec[iv + 1] - tvec[iv]) * (1.0f / (float)KSUB_);
    for (int s = 0; s < KSUB_; ++s) {
#pragma unroll
      for (int st = 0; st < 6; ++st) {
        // stage input (per-thread, RK coefficients folded to immediates)
        float acc = 0.0f;
#pragma unroll
        for (int j = 0; j < st; ++j) acc += RK_A[st][j] * kreg[j];
        zs[tid] = zreg + h * acc;
        __syncthreads();

        // layer 1: (16x16)@(16x128)+b1, silu
        v8f c = {};
#pragma unroll
        for (int c2 = 0; c2 < 4; ++c2) {
          v2f a = *(const v2f*)(zs + r * 16 + 4 * c2 + 2 * half);
          c = wmma_acc(a, w1f[c2], c);
        }
#pragma unroll
        for (int i = 0; i < 8; ++i)
          a1[(i + 8 * half) * 128 + n0 + r] = silu_f(c[i] + bias1);
        __syncthreads();

        // layer 2: (16x128)@(128x128)+b2, silu
        c = (v8f){};
#pragma unroll
        for (int c2 = 0; c2 < 32; ++c2) {
          v2f a = *(const v2f*)(a1 + r * 128 + 4 * c2 + 2 * half);
          c = wmma_acc(a, w2f[c2], c);
        }
#pragma unroll
        for (int i = 0; i < 8; ++i)
          a2[(i + 8 * half) * 128 + n0 + r] = silu_f(c[i] + bias2);
        __syncthreads();

        // layer 3 partials: wave w covers K in [16w,16w+16)
        c = (v8f){};
#pragma unroll
        for (int c2 = 0; c2 < 4; ++c2) {
          v2f a = *(const v2f*)(a2 + r * 128 + n0 + 4 * c2 + 2 * half);
          c = wmma_acc(a, w3f[c2], c);
        }
#pragma unroll
        for (int i = 0; i < 8; ++i)
          part[w * 256 + (i + 8 * half) * 16 + r] = c[i];
        __syncthreads();

        // K-reduce + bias (thread t owns element t)
        float sv = bias3;
#pragma unroll
        for (int ww = 0; ww < 8; ++ww) sv += part[ww * 256 + tid];

        // layernorm over D=16: rows are 16 consecutive lanes -> shfl butterflies
        float m = sv;
#pragma unroll
        for (int off = 1; off < 16; off <<= 1) m += __shfl_xor(m, off, 16);
        m *= (1.0f / 16.0f);
        const float d = sv - m;
        float vv = d * d;
#pragma unroll
        for (int off = 1; off < 16; off <<= 1) vv += __shfl_xor(vv, off, 16);
        vv *= (1.0f / 16.0f);
        kreg[st] = d * __builtin_amdgcn_rsqf(vv + LN_EPS_) * gam + bet;
      }
      zreg += h * (RK_B[0] * kreg[0] + RK_B[2] * kreg[2] + RK_B[3] * kreg[3] +
                   RK_B[4] * kreg[4] + RK_B[5] * kreg[5]);
    }
    ztraj[((g * 16 + row) * L_ + iv + 1) * D_ + col] = zreg;
  }
}

__launch_bounds__(256) __global__
void enc_kernel(const float* __restrict__ x_seq, const float* __restrict__ eps,
                const float* __restrict__ ew1, const float* __restrict__ eb1,
                const float* __restrict__ ew2, const float* __restrict__ eb2,
                const float* __restrict__ ew3, const float* __restrict__ eb3,
                const float* __restrict__ muw, const float* __restrict__ mub,
                const float* __restrict__ lvw, const float* __restrict__ lvb,
                float* __restrict__ out_mu, float* __restrict__ out_lv,
                float* __restrict__ z0buf, float* __restrict__ ztraj) {
  extern __shared__ float smem[];
  float* x0 = smem;       // 16*256
  float* h1 = x0 + 4096;  // 16*512
  float* h2 = h1 + 8192;  // 16*256
  float* h3 = h2 + 4096;  // 16*128
  float* pm = h3 + 2048;  // 8*256
  float* pv = pm + 2048;  // 8*256

  const int g = blockIdx.x, tid = threadIdx.x;
  const int w = tid >> 5, lane = tid & 31, half = lane >> 4, r = lane & 15;

  for (int i = tid; i < 16 * 256; i += 256) {
    const int row = i >> 8, col = i & 255;
    x0[i] = x_seq[((g * 16 + row) * L_) * N_ + col];  // l = 0
  }
  __syncthreads();

  for (int j = 0; j < 4; ++j) {  // e1: (16x256)@(256x512), relu
    const int n0 = (w * 4 + j) * 16;
    v8f c = {};
    for (int k0 = 0; k0 < 256; k0 += 4)
      c = wmma4(x0, 256, k0, ew1, 512, n0, c, r, half);
    const float bias = eb1[n0 + r];
#pragma unroll
    for (int i = 0; i < 8; ++i) {
      const float v = c[i] + bias;
      h1[(i + 8 * half) * 512 + n0 + r] = v > 0.f ? v : 0.f;
    }
  }
  __syncthreads();
  for (int j = 0; j < 2; ++j) {  // e2: (16x512)@(512x256), relu
    const int n0 = (w * 2 + j) * 16;
    v8f c = {};
    for (int k0 = 0; k0 < 512; k0 += 4)
      c = wmma4(h1, 512, k0, ew2, 256, n0, c, r, half);
    const float bias = eb2[n0 + r];
#pragma unroll
    for (int i = 0; i < 8; ++i) {
      const float v = c[i] + bias;
      h2[(i + 8 * half) * 256 + n0 + r] = v > 0.f ? v : 0.f;
    }
  }
  __syncthreads();
  {  // e3: (16x256)@(256x128), relu
    const int n0 = w * 16;
    v8f c = {};
    for (int k0 = 0; k0 < 256; k0 += 4)
      c = wmma4(h2, 256, k0, ew3, 128, n0, c, r, half);
    const float bias = eb3[n0 + r];
#pragma unroll
    for (int i = 0; i < 8; ++i) {
      const float v = c[i] + bias;
      h3[(i + 8 * half) * 128 + n0 + r] = v > 0.f ? v : 0.f;
    }
  }
  __syncthreads();
  {  // mu / logvar heads: K=128 split across waves
    v8f cm = {}, cv = {};
#pragma unroll
    for (int kk = 0; kk < 16; kk += 4) {
      cm = wmma4(h3, 128, w * 16 + kk, muw, 16, 0, cm, r, half);
      cv = wmma4(h3, 128, w * 16 + kk, lvw, 16, 0, cv, r, half);
    }
#pragma unroll
    for (int i = 0; i < 8; ++i) {
      pm[w * 256 + (i + 8 * half) * 16 + r] = cm[i];
      pv[w * 256 + (i + 8 * half) * 16 + r] = cv[i];
    }
  }
  __syncthreads();
  {
    const int row = tid >> 4, col = tid & 15;
    const int bg = g * 16 + row;
    float smu = mub[col], slv = lvb[col];
#pragma unroll
    for (int ww = 0; ww < 8; ++ww) {
      smu += pm[ww * 256 + tid];
      slv += pv[ww * 256 + tid];
    }
    out_mu[bg * 16 + col] = smu;
    out_lv[bg * 16 + col] = slv;
    const float zz = smu + expf(0.5f * slv) * eps[bg * 16 + col];
    z0buf[bg * 16 + col] = zz;
    ztraj[(bg * L_) * D_ + col] = zz;  // z_traj[:,0,:]
  }
}

// Decoder: 64 rows (4 row-tiles) per workgroup so one weight fragment feeds 4 wmma.
__launch_bounds__(256) __global__
void dec_kernel(const float* __restrict__ ztraj, const float* __restrict__ dw1,
                const float* __restrict__ db1, const float* __restrict__ dw2,
                const float* __restrict__ db2, const float* __restrict__ dw3,
                const float* __restrict__ db3, float* __restrict__ xhat) {
  extern __shared__ float smem[];
  float* zt = smem;                // ROWT_*16*16
  float* h1 = zt + ROWT_ * 256;    // ROWT_*16*512
  float* h2 = h1 + ROWT_ * 8192;   // ROWT_*16*512

  const int bt = blockIdx.x, tid = threadIdx.x;
  const int w = tid >> 5, lane = tid & 31, half = lane >> 4, r = lane & 15;
  const int row0 = bt * (16 * ROWT_);

  for (int i = tid; i < ROWT_ * 256; i += 256) {
    const int row = i >> 4, col = i & 15;
    zt[i] = ztraj[(row0 + row) * 16 + col];
  }
  __syncthreads();

  // d1: (64x16)@(16x512), relu
  for (int j = 0; j < 4; ++j) {
    const int n0 = (w * 4 + j) * 16;
    v8f c[ROWT_] = {};
#pragma unroll
    for (int k0 = 0; k0 < 16; k0 += 4) {
      const int kk = k0 + 2 * half;
      v2f b;
      b.x = dw1[kk * 512 + n0 + r];
      b.y = dw1[(kk + 1) * 512 + n0 + r];
#pragma unroll
      for (int t2 = 0; t2 < ROWT_; ++t2) {
        v2f a = *(const v2f*)(zt + t2 * 256 + r * 16 + kk);
        c[t2] = wmma_acc(a, b, c[t2]);
      }
    }
    const float bias = db1[n0 + r];
#pragma unroll
    for (int t2 = 0; t2 < ROWT_; ++t2)
#pragma unroll
      for (int i = 0; i < 8; ++i) {
        const float v = c[t2][i] + bias;
        h1[t2 * 8192 + (i + 8 * half) * 512 + n0 + r] = v > 0.f ? v : 0.f;
      }
  }
  __syncthreads();
  // d2: (64x512)@(512x512), relu
  for (int j = 0; j < 4; ++j) {
    const int n0 = (w * 4 + j) * 16;
    v8f c[ROWT_] = {};
    for (int k0 = 0; k0 < 512; k0 += 4) {
      const int kk = k0 + 2 * half;
      v2f b;
      b.x = dw2[kk * 512 + n0 + r];
      b.y = dw2[(kk + 1) * 512 + n0 + r];
#pragma unroll
      for (int t2 = 0; t2 < ROWT_; ++t2) {
        v2f a = *(const v2f*)(h1 + t2 * 8192 + r * 512 + kk);
        c[t2] = wmma_acc(a, b, c[t2]);
      }
    }
    const float bias = db2[n0 + r];
#pragma unroll
    for (int t2 = 0; t2 < ROWT_; ++t2)
#pragma unroll
      for (int i = 0; i < 8; ++i) {
        const float v = c[t2][i] + bias;
        h2[t2 * 8192 + (i + 8 * half) * 512 + n0 + r] = v > 0.f ? v : 0.f;
      }
  }
  __syncthreads();
  // d3: (64x512)@(512x256) + bias -> xhat
  for (int j = 0; j < 2; ++j) {
    const int n0 = (w * 2 + j) * 16;
    v8f c[ROWT_] = {};
    for (int k0 = 0; k0 < 512; k0 += 4) {
      const int kk = k0 + 2 * half;
      v2f b;
      b.x = dw3[kk * 256 + n0 + r];
      b.y = dw3[(kk + 1) * 256 + n0 + r];
#pragma unroll
      for (int t2 = 0; t2 < ROWT_; ++t2) {
        v2f a = *(const v2f*)(h2 + t2 * 8192 + r * 512 + kk);
        c[t2] = wmma_acc(a, b, c[t2]);
      }
    }
    const float bias = db3[n0 + r];
#pragma unroll
    for (int t2 = 0; t2 < ROWT_; ++t2)
#pragma unroll
      for (int i = 0; i < 8; ++i)
        xhat[(row0 + t2 * 16 + i + 8 * half) * 256 + n0 + r] = c[t2][i] + bias;
  }
}

__global__ void zdiff_kernel(const float* __restrict__ ztraj,
                             const float* __restrict__ tvec,
                             float* __restrict__ zdiff) {
  const int idx = blockIdx.x * blockDim.x + threadIdx.x;
  if (idx >= B_ * (L_ - 1) * D_) return;
  const int d = idx & 15;
  const int rem = idx >> 4;  // b*(L-1) + i
  const int i = rem % (L_ - 1);
  const int b = rem / (L_ - 1);
  const float dt = tvec[i + 1] - tvec[i];
  zdiff[idx] =
      (ztraj[(b * L_ + i + 1) * D_ + d] - ztraj[(b * L_ + i) * D_ + d]) / dt;
}

extern "C" void kernel_launch(void* const* d_in, const int* in_sizes, int n_in,
                              void* d_out, int out_size, void* d_ws,
                              size_t ws_size, hipStream_t stream) {
  const float* x_seq = (const float*)d_in[0];
  const float* tvec = (const float*)d_in[1];
  const float* eps = (const float*)d_in[2];
  const float* ew1 = (const float*)d_in[3];
  const float* eb1 = (const float*)d_in[4];
  const float* ew2 = (const float*)d_in[5];
  const float* eb2 = (const float*)d_in[6];
  const float* ew3 = (const float*)d_in[7];
  const float* eb3 = (const float*)d_in[8];
  const float* muw = (const float*)d_in[9];
  const float* mub = (const float*)d_in[10];
  const float* lvw = (const float*)d_in[11];
  const float* lvb = (const float*)d_in[12];
  const float* ow1 = (const float*)d_in[13];
  const float* ob1 = (const float*)d_in[14];
  const float* ow2 = (const float*)d_in[15];
  const float* ob2 = (const float*)d_in[16];
  const float* ow3 = (const float*)d_in[17];
  const float* ob3 = (const float*)d_in[18];
  const float* lng = (const float*)d_in[19];
  const float* lnb = (const float*)d_in[20];
  const float* dw1 = (const float*)d_in[21];
  const float* db1 = (const float*)d_in[22];
  const float* dw2 = (const float*)d_in[23];
  const float* db2 = (const float*)d_in[24];
  const float* dw3 = (const float*)d_in[25];
  const float* db3 = (const float*)d_in[26];

  float* out = (float*)d_out;
  float* xhat = out;                            // B*L*N
  float* out_mu = xhat + (size_t)B_ * L_ * N_;  // B*D
  float* out_lv = out_mu + B_ * D_;             // B*D
  float* ztraj = out_lv + B_ * D_;              // B*L*D
  float* zdiff = ztraj + (size_t)B_ * L_ * D_;  // B*(L-1)*D

  float* z0buf = (float*)d_ws;  // B*D floats

  const size_t enc_smem =
      (size_t)(4096 + 8192 + 4096 + 2048 + 2048 + 2048) * sizeof(float);
  enc_kernel<<<B_ / 16, 256, enc_smem, stream>>>(
      x_seq, eps, ew1, eb1, ew2, eb2, ew3, eb3, muw, mub, lvw, lvb, out_mu,
      out_lv, z0buf, ztraj);

  const size_t ode_smem = (size_t)(256 + 2048 + 2048 + 2048) * sizeof(float);
  ode_kernel<<<B_ / 16, 256, ode_smem, stream>>>(tvec, ow1, ob1, ow2, ob2, ow3,
                                                 ob3, lng, lnb, z0buf, ztraj);

  const size_t dec_smem =
      (size_t)(ROWT_ * 256 + ROWT_ * 8192 + ROWT_ * 8192) * sizeof(float);
  dec_kernel<<<(B_ * L_) / (16 * ROWT_), 256, dec_smem, stream>>>(
      ztraj, dw1, db1, dw2, db2, dw3, db3, xhat);

  const int nzd = B_ * (L_ - 1) * D_;
  zdiff_kernel<<<(nzd + 255) / 256, 256, 0, stream>>>(ztraj, tvec, zdiff);
}